// DBSCAN_32358283608448
// MI455X (gfx1250) — compile-verified
//
#include <hip/hip_runtime.h>
#include <stdint.h>

// ---------------- problem constants ----------------
#define NPTS   12288
#define DIMS   8
#define NT     (NPTS / 16)      // 768 16x16 tiles per side
#define NW     (NPTS / 32)      // 384 adjacency words per row
#define EPS2   0.25f
#define MINS   5
#define SENT   NPTS
#define CHUNKS 8                // column chunks per row-tile (768/8 = 96 J-tiles each)

typedef __attribute__((ext_vector_type(2))) float v2f;
typedef __attribute__((ext_vector_type(8))) float v8f;

// ---------------- 1) squared norms + density zero ----------------
__global__ void k_pre(const float* __restrict__ X, float* __restrict__ sq,
                      int* __restrict__ density) {
    int i = blockIdx.x * blockDim.x + threadIdx.x;
    if (i < NPTS) {
        float s = 0.f;
#pragma unroll
        for (int d = 0; d < DIMS; ++d) { float v = X[i * DIMS + d]; s += v * v; }
        sq[i] = s;
        density[i] = 0;
    }
}

// ---------------- 2) WMMA distance tiles -> adjacency bitset + density ----------
// One wave handles one 16-row strip (rowTile) over a chunk of 96 J-tiles.
// A-fragment (fp32 16x4, ISA layout): lanes 0-15 hold {K=0,K=1}, lanes 16-31 {K=2,K=3}
// for row M = lane&15.  B-fragment mirrors this with N = lane&15.
__global__ void k_tiles(const float* __restrict__ X, const float* __restrict__ sq,
                        unsigned* __restrict__ adj, int* __restrict__ density) {
    const int lane    = threadIdx.x & 31;
    const int wave    = threadIdx.x >> 5;
    const int rowTile = blockIdx.x * 8 + wave;            // 0..767
    const int chunk   = blockIdx.y;                        // 0..7
    const int rowBase = rowTile * 16;
    const int hi      = (lane >= 16) ? 1 : 0;
    const int rsub    = lane & 15;

    // A fragments (K=0..3 and K=4..7), reused for every J-tile
    const int arow = rowBase + rsub;
    v2f a0 = *(const v2f*)(X + arow * DIMS + hi * 2);       // K 0,1 / 2,3
    v2f a1 = *(const v2f*)(X + arow * DIMS + 4 + hi * 2);   // K 4,5 / 6,7

    // per-lane squared norms of the 8 C-register rows (M = v + 8*hi)
    float srow[8];
#pragma unroll
    for (int v = 0; v < 8; ++v) srow[v] = sq[rowBase + v + 8 * hi];

    unsigned pc = 0;  // popcount accumulator (valid for lanes 0..15, row = lane)

    const int jt0 = chunk * (NT / CHUNKS);
    for (int jt = jt0; jt < jt0 + (NT / CHUNKS); jt += 2) {
        const int col0 = jt * 16 + rsub;
        const int col1 = col0 + 16;
        v2f b00 = *(const v2f*)(X + col0 * DIMS + hi * 2);
        v2f b01 = *(const v2f*)(X + col0 * DIMS + 4 + hi * 2);
        v2f b10 = *(const v2f*)(X + col1 * DIMS + hi * 2);
        v2f b11 = *(const v2f*)(X + col1 * DIMS + 4 + hi * 2);

        v8f c0 = {}, c1 = {};
        c0 = __builtin_amdgcn_wmma_f32_16x16x4_f32(false, a0, false, b00, (short)0, c0, false, false);
        c0 = __builtin_amdgcn_wmma_f32_16x16x4_f32(false, a1, false, b01, (short)0, c0, false, false);
        c1 = __builtin_amdgcn_wmma_f32_16x16x4_f32(false, a0, false, b10, (short)0, c1, false, false);
        c1 = __builtin_amdgcn_wmma_f32_16x16x4_f32(false, a1, false, b11, (short)0, c1, false, false);

        const float sc0 = sq[col0];
        const float sc1 = sq[col1];

        unsigned bits0 = 0, bits1 = 0;   // 16 adjacency bits for "my" row
#pragma unroll
        for (int v = 0; v < 8; ++v) {
            // C layout: VGPR v, lanes 0-15 -> row v, lanes 16-31 -> row v+8
            unsigned m0 = __builtin_amdgcn_ballot_w32((srow[v] + sc0 - 2.0f * c0[v]) <= EPS2);
            unsigned m1 = __builtin_amdgcn_ballot_w32((srow[v] + sc1 - 2.0f * c1[v]) <= EPS2);
            if (lane == v)     { bits0 = m0 & 0xffffu; bits1 = m1 & 0xffffu; }
            if (lane == v + 8) { bits0 = m0 >> 16;     bits1 = m1 >> 16;     }
        }
        if (lane < 16) {
            unsigned word = bits0 | (bits1 << 16);     // cols [jt*16, jt*16+32)
            adj[(size_t)(rowBase + lane) * NW + (jt >> 1)] = word;
            pc += __popc(word);
        }
    }
    if (lane < 16) atomicAdd(&density[rowBase + lane], (int)pc);
}

// ---------------- 3) core mask + label init ----------------
__global__ void k_core(const int* __restrict__ density, unsigned* __restrict__ coreMask,
                       int* __restrict__ lbl) {
    int i = blockIdx.x * blockDim.x + threadIdx.x;
    bool c = density[i] >= MINS;
    unsigned m = __builtin_amdgcn_ballot_w32(c);
    if ((threadIdx.x & 31) == 0) coreMask[i >> 5] = m;
    lbl[i] = c ? i : SENT;
}

// ---------------- 4) min-label sweep over core-core edges (wave per point) -----
__global__ void k_sweep(const int* __restrict__ lin, int* __restrict__ lout,
                        const unsigned* __restrict__ adj,
                        const unsigned* __restrict__ coreMask) {
    const int lane = threadIdx.x & 31;
    const int i    = (blockIdx.x * blockDim.x + threadIdx.x) >> 5;   // exact: grid covers NPTS waves
    const bool isCore = (coreMask[i >> 5] >> (i & 31)) & 1u;
    int m = SENT;
    if (isCore) {
        m = lin[i];
        const unsigned* row = adj + (size_t)i * NW;
        for (int w = lane; w < NW; w += 32) {
            unsigned bits = row[w] & coreMask[w];
            while (bits) {
                int b = __builtin_ctz(bits);
                bits &= bits - 1;
                int lj = lin[w * 32 + b];
                m = lj < m ? lj : m;
            }
        }
    }
#pragma unroll
    for (int off = 16; off; off >>= 1) {
        int o = __shfl_xor(m, off, 32);
        m = o < m ? o : m;
    }
    if (lane == 0) lout[i] = m;
}

// ---------------- 5) pointer jump (path halving) ----------------
__global__ void k_jump(const int* __restrict__ lin, int* __restrict__ lout) {
    int i = blockIdx.x * blockDim.x + threadIdx.x;
    int l = lin[i];
    lout[i] = (l < SENT) ? lin[l] : SENT;
}

// ---------------- 6) border assignment ----------------
__global__ void k_border(const int* __restrict__ lbl, int* __restrict__ root,
                         const unsigned* __restrict__ adj,
                         const unsigned* __restrict__ coreMask) {
    const int lane = threadIdx.x & 31;
    const int i    = (blockIdx.x * blockDim.x + threadIdx.x) >> 5;
    const bool isCore = (coreMask[i >> 5] >> (i & 31)) & 1u;
    if (isCore) {                       // wave-uniform branch
        if (lane == 0) root[i] = lbl[i];
        return;
    }
    int m = SENT;
    const unsigned* row = adj + (size_t)i * NW;
    for (int w = lane; w < NW; w += 32) {
        unsigned bits = row[w] & coreMask[w];
        while (bits) {
            int b = __builtin_ctz(bits);
            bits &= bits - 1;
            int lj = lbl[w * 32 + b];
            m = lj < m ? lj : m;
        }
    }
#pragma unroll
    for (int off = 16; off; off >>= 1) {
        int o = __shfl_xor(m, off, 32);
        m = o < m ? o : m;
    }
    if (lane == 0) root[i] = m;
}

// ---------------- 7) rank = cumsum(is_root) - 1  (single block) ----------------
__global__ void k_scan(const int* __restrict__ lbl, const unsigned* __restrict__ coreMask,
                       int* __restrict__ rank) {
    __shared__ int buf[1024];
    __shared__ int carry;
    if (threadIdx.x == 0) carry = 0;
    __syncthreads();
    for (int base = 0; base < NPTS; base += 1024) {
        int i = base + threadIdx.x;
        bool isCore = (coreMask[i >> 5] >> (i & 31)) & 1u;
        buf[threadIdx.x] = (isCore && lbl[i] == i) ? 1 : 0;
        __syncthreads();
        for (int off = 1; off < 1024; off <<= 1) {
            int t = (threadIdx.x >= off) ? buf[threadIdx.x - off] : 0;
            __syncthreads();
            buf[threadIdx.x] += t;
            __syncthreads();
        }
        rank[i] = carry + buf[threadIdx.x] - 1;
        __syncthreads();
        if (threadIdx.x == 1023) carry += buf[1023];
        __syncthreads();
    }
}

// ---------------- 8) final labels ----------------
__global__ void k_final(const int* __restrict__ root, const int* __restrict__ rank,
                        int* __restrict__ out) {
    int i = blockIdx.x * blockDim.x + threadIdx.x;
    int r = root[i];
    out[i] = (r < SENT) ? rank[r] : -1;
}

// ---------------- launch ----------------
extern "C" void kernel_launch(void* const* d_in, const int* in_sizes, int n_in,
                              void* d_out, int out_size, void* d_ws, size_t ws_size,
                              hipStream_t stream) {
    const float* X = (const float*)d_in[0];
    int* out = (int*)d_out;   // reference output dtype is int32

    char* ws = (char*)d_ws;
    float*    sq       = (float*)(ws);
    int*      density  = (int*)(ws + (size_t)NPTS * 4 * 1);
    int*      lbl0     = (int*)(ws + (size_t)NPTS * 4 * 2);
    int*      lbl1     = (int*)(ws + (size_t)NPTS * 4 * 3);
    int*      root     = (int*)(ws + (size_t)NPTS * 4 * 4);
    int*      rank     = (int*)(ws + (size_t)NPTS * 4 * 5);
    unsigned* coreMask = (unsigned*)(ws + (size_t)NPTS * 4 * 6);            // NW words
    unsigned* adj      = (unsigned*)(ws + (size_t)NPTS * 4 * 6 + 2048);     // NPTS*NW words (~18.9 MB)

    const int B = 256;
    const int GB = NPTS / B;               // 48

    k_pre<<<GB, B, 0, stream>>>(X, sq, density);

    dim3 tg(NT / 8, CHUNKS);               // 96 x 8 blocks of 8 waves
    k_tiles<<<tg, 256, 0, stream>>>(X, sq, adj, density);

    k_core<<<GB, B, 0, stream>>>(density, coreMask, lbl0);

    const int WAVES_PER_BLK = 8;           // 256 threads
    const int GW = NPTS / WAVES_PER_BLK;   // 1536 blocks, one wave per point

    // 32 rounds of (neighbor-min sweep + 3 pointer jumps); net lbl0 -> lbl0.
    for (int it = 0; it < 32; ++it) {
        k_sweep<<<GW, 256, 0, stream>>>(lbl0, lbl1, adj, coreMask);
        k_jump <<<GB, B,   0, stream>>>(lbl1, lbl0);
        k_jump <<<GB, B,   0, stream>>>(lbl0, lbl1);
        k_jump <<<GB, B,   0, stream>>>(lbl1, lbl0);
    }

    k_border<<<GW, 256, 0, stream>>>(lbl0, root, adj, coreMask);
    k_scan  <<<1, 1024, 0, stream>>>(lbl0, coreMask, rank);
    k_final <<<GB, B,   0, stream>>>(root, rank, out);
}